// VectorQuantizer_47562467836174
// MI455X (gfx1250) — compile-verified
//
#include <hip/hip_runtime.h>

typedef __attribute__((ext_vector_type(2))) float v2f;
typedef __attribute__((ext_vector_type(4))) float v4f;
typedef __attribute__((ext_vector_type(8))) float v8f;

#define K_CODES 512
#define D_CH    64
#define HW      4096            // 64*64 spatial per image
#define WAVES   8               // waves per block (wave32)
#define POS_PER_WAVE 16
#define POS_PER_BLOCK (WAVES * POS_PER_WAVE)   // 128
#define N_POS   131072          // 32*64*64
#define NBLOCKS (N_POS / POS_PER_BLOCK)        // 1024

// ---------------------------------------------------------------------------
// Kernel 0: ||w_k||^2 for all 512 codes -> d_ws[1024 + k]
// ---------------------------------------------------------------------------
__global__ __launch_bounds__(256)
void vq_wnorm_kernel(const float* __restrict__ w, float* __restrict__ wnorm) {
    int c = blockIdx.x * blockDim.x + threadIdx.x;
    if (c < K_CODES) {
        const v4f* row = (const v4f*)(w + c * D_CH);
        float s = 0.f;
#pragma unroll
        for (int i = 0; i < D_CH / 4; ++i) {
            v4f v = row[i];
            s += v.x * v.x + v.y * v.y + v.z * v.z + v.w * v.w;
        }
        wnorm[c] = s;
    }
}

// ---------------------------------------------------------------------------
// Kernel 1: nearest-code search with V_WMMA_F32_16X16X4_F32, quantized
// scatter back to NCHW, per-block loss partial -> partials[blockIdx.x]
// ---------------------------------------------------------------------------
__global__ __launch_bounds__(256)
void vq_main_kernel(const float* __restrict__ x,     // [32,64,64,64] NCHW
                    const float* __restrict__ w,     // [512,64]
                    const float* __restrict__ wnorm, // [512]
                    float* __restrict__ out,         // [32,64,64,64] NCHW
                    float* __restrict__ partials) {  // [NBLOCKS]
    __shared__ int   idx_lds[WAVES * 16];
    __shared__ float wavesum[WAVES];

    const int tid    = threadIdx.x;
    const int lane   = tid & 31;
    const int wv     = tid >> 5;
    const int laneLo = lane & 15;      // M (position within tile) / N (code within tile)
    const int laneHi = lane >> 4;      // selects K pair within WMMA fragments

    // 16 consecutive flattened positions per wave; 128 | 4096 so same batch b.
    const int posBase = blockIdx.x * POS_PER_BLOCK + wv * POS_PER_WAVE;
    const int b   = posBase >> 12;             // /4096
    const int hw  = (posBase & 4095) + laneLo; // this lane's spatial index
    const float* xb = x + (size_t)b * D_CH * HW + hw;

    // ---- Load A fragments: x tile [16 pos x 64 ch] as 16 chunks of 16x4 ----
    // lane L holds (M = L%16, K = 2*(L/16) + {0,1}) per ISA A-layout.
    v2f a[16];
#pragma unroll
    for (int t = 0; t < 16; ++t) {
        int ch = t * 4 + laneHi * 2;
        a[t].x = xb[(size_t)ch * HW];
        a[t].y = xb[(size_t)(ch + 1) * HW];
    }

    // ---- Running argmin of (||w||^2 - 2 x.w) over 32 groups of 16 codes ----
    float best[8];
    int   bidx[8];
#pragma unroll
    for (int j = 0; j < 8; ++j) { best[j] = 3.4e38f; bidx[j] = 0; }

    for (int ct = 0; ct < K_CODES / 16; ++ct) {
        const int code = ct * 16 + laneLo;               // N-index this lane covers
        const float* wrow = w + code * D_CH + laneHi * 2; // 8-byte aligned
        v8f acc = {};
#pragma unroll
        for (int t = 0; t < 16; ++t) {
            // B fragment 4x16: lane L holds (K = 2*(L/16)+{0,1}, N = L%16)
            v2f bf = *(const v2f*)(wrow + t * 4);
            acc = __builtin_amdgcn_wmma_f32_16x16x4_f32(
                false, a[t], false, bf, (short)0, acc, false, false);
        }
        const float wn = wnorm[code];
#pragma unroll
        for (int j = 0; j < 8; ++j) {   // acc[j]: M = j + 8*laneHi, N = laneLo
            float s = wn - 2.0f * acc[j];
            if (s < best[j]) { best[j] = s; bidx[j] = code; }  // strict <: first-min wins
        }
    }

    // ---- Cross-lane argmin within each 16-lane half (rows stay in half) ----
#pragma unroll
    for (int j = 0; j < 8; ++j) {
        float bs = best[j]; int bi = bidx[j];
#pragma unroll
        for (int off = 1; off < 16; off <<= 1) {
            float os = __shfl_xor(bs, off, 32);
            int   oi = __shfl_xor(bi, off, 32);
            if (os < bs || (os == bs && oi < bi)) { bs = os; bi = oi; }
        }
        best[j] = bs; bidx[j] = bi;
    }
    // lanes 0 and 16 publish rows 0-7 / 8-15 (same-wave LDS ops are in-order)
    if (laneLo == 0) {
#pragma unroll
        for (int j = 0; j < 8; ++j) idx_lds[wv * 16 + laneHi * 8 + j] = bidx[j];
    }

    const int myIdx = idx_lds[wv * 16 + laneLo];
    const float* wq = w + (size_t)myIdx * D_CH;

    // ---- Loss partial: lanes L and L+16 cover disjoint channel pairs ----
    float lsum = 0.f;
#pragma unroll
    for (int t = 0; t < 16; ++t) {
        int ch = t * 4 + laneHi * 2;
        float d0 = wq[ch]     - a[t].x;
        float d1 = wq[ch + 1] - a[t].y;
        lsum += d0 * d0 + d1 * d1;
    }
#pragma unroll
    for (int off = 1; off < 32; off <<= 1) lsum += __shfl_xor(lsum, off, 32);
    if (lane == 0) wavesum[wv] = lsum;

    // ---- Scatter quantized vectors back to NCHW (coalesced per half-wave) --
    float* ob = out + (size_t)b * D_CH * HW + hw;
#pragma unroll
    for (int t = 0; t < 32; ++t) {
        int ch = t * 2 + laneHi;       // half 0: even channels, half 1: odd
        ob[(size_t)ch * HW] = wq[ch];
    }

    __syncthreads();
    if (tid == 0) {
        float s = 0.f;
#pragma unroll
        for (int i = 0; i < WAVES; ++i) s += wavesum[i];
        partials[blockIdx.x] = s;
    }
}

// ---------------------------------------------------------------------------
// Kernel 2: deterministic reduce of block partials -> scalar loss
// loss = 1.25 * sum / (N*D)
// ---------------------------------------------------------------------------
__global__ __launch_bounds__(256)
void vq_finalize_kernel(const float* __restrict__ partials, int n,
                        float* __restrict__ loss_out, float scale) {
    __shared__ float red[256];
    float s = 0.f;
    for (int i = threadIdx.x; i < n; i += 256) s += partials[i];
    red[threadIdx.x] = s;
    __syncthreads();
    for (int st = 128; st > 0; st >>= 1) {
        if ((int)threadIdx.x < st) red[threadIdx.x] += red[threadIdx.x + st];
        __syncthreads();
    }
    if (threadIdx.x == 0) loss_out[0] = red[0] * scale;
}

// ---------------------------------------------------------------------------
extern "C" void kernel_launch(void* const* d_in, const int* in_sizes, int n_in,
                              void* d_out, int out_size, void* d_ws, size_t ws_size,
                              hipStream_t stream) {
    (void)in_sizes; (void)n_in; (void)ws_size;
    const float* x = (const float*)d_in[0];   // [32,64,64,64] fp32
    const float* w = (const float*)d_in[1];   // [512,64] fp32
    float* out      = (float*)d_out;          // [32,64,64,64] then loss scalar
    float* partials = (float*)d_ws;           // [0 .. NBLOCKS)
    float* wnorm    = partials + NBLOCKS;     // [NBLOCKS .. NBLOCKS+512)

    vq_wnorm_kernel<<<(K_CODES + 255) / 256, 256, 0, stream>>>(w, wnorm);
    vq_main_kernel<<<NBLOCKS, 256, 0, stream>>>(x, w, wnorm, out, partials);
    vq_finalize_kernel<<<1, 256, 0, stream>>>(
        partials, NBLOCKS, out + (out_size - 1),
        1.25f / (float)(N_POS * D_CH));
}